// SL_CSC_FISTA_26740466385537
// MI455X (gfx1250) — compile-verified
//
#include <hip/hip_runtime.h>
#include <cmath>

// ---------------------------------------------------------------------------
// SL-CSC FISTA on MI455X (gfx1250), fp32 WMMA (V_WMMA_F32_16X16X4_F32).
//  - Dt(R): direct GEMM [64 atoms x 84(K,pad)] x [84 x 16 cols], operands
//           LDS-staged per output row -> branch-free WMMA inner loop.
//  - D(Xu): Toeplitz-banded GEMM, 16x32 output-pixel tiles per wave (2 WMMA
//           share each B fetch), soft-threshold fused into the A-patch load.
// Workspace requirement: ~120 MB (g + 3 X-sized buffers + small scratch).
// ---------------------------------------------------------------------------

typedef __attribute__((ext_vector_type(2))) float v2f;
typedef __attribute__((ext_vector_type(8))) float v8f;

#define TAUC 0.1f
#define NB 8
#define NA 64
#define HY 128
#define WY 128
#define HX 120
#define WX 120
#define NXEL (NB*NA*HX*WX)   // 7,372,800
#define NYEL (NB*HY*WY)      // 131,072
#define L1BLOCKS 1024
#define DTILES   256         // 8 b * 8 y0 * 4 x0-pairs

// ---------------- weight prep -------------------------------------------------
// Wp: packed Dt weights [4 atom-tiles][84 K (padded)][16 m]
__global__ void prep_dt_kernel(const float* __restrict__ Wd, float* __restrict__ Wp) {
  int tid = blockIdx.x * blockDim.x + threadIdx.x;
  if (tid >= 4 * 84 * 16) return;
  int m  = tid & 15;
  int kk = (tid >> 4) % 84;
  int at = tid / (84 * 16);
  int a  = at * 16 + m;
  float val = 0.f;
  if (kk < 81) {
    int u = kk / 9, v = kk % 9;
    val = Wd[a * 81 + v * 9 + u];        // Wt[a,u,v] = Wd[a,0,v,u]
  }
  Wp[tid] = val;
}

// Btp: Toeplitz-banded D weights, lane-packed:
//   Btp[((a*9+u)*6 + c)*64 + vg*32 + lane]
//   value = Wf[a,u,t-n], t = 4c + (lane>>4)*2 + vg, n = lane&15, band 0<=t-n<9
//   Wf[a,u,v] = Wd[a,0,8-u,8-v]  (flipped kernel of conv_transpose)
__global__ void prep_toep_kernel(const float* __restrict__ Wd, float* __restrict__ Btp) {
  int tid = blockIdx.x * blockDim.x + threadIdx.x;   // exactly 64*9*6*64 = 221184
  int lane = tid & 31;
  int idx  = tid >> 5;
  int vg   = idx & 1;  idx >>= 1;
  int c    = idx % 6;  idx /= 6;
  int u    = idx % 9;
  int a    = idx / 9;
  if (a >= NA) return;
  int t = 4 * c + ((lane >> 4) << 1) + vg;
  int n = lane & 15;
  int v = t - n;
  float val = 0.f;
  if (v >= 0 && v < 9) val = Wd[a * 81 + (8 - u) * 9 + (8 - v)];
  Btp[tid] = val;
}

// ---------------- Dt: g = Dt(R) ----------------------------------------------
// grid (120 rows, 8 batch), block 128 = 4 waves, wave w -> atom tile w.
// All operands staged in LDS once; inner loop is pure ds_load + WMMA.
__global__ __launch_bounds__(128) void dt_wmma_kernel(const float* __restrict__ R,
                                                      const float* __restrict__ Wp,
                                                      float* __restrict__ g) {
  __shared__ float Rt[10 * 144];     // rows i..i+8 (cols 0..127, zero-pad to 144)
                                     // row 9 = all zeros (K-padding taps)
  __shared__ float Wsh[4 * 84 * 16]; // packed weights

  int i = blockIdx.x;
  int b = blockIdx.y;
  int tid   = threadIdx.x;
  int lane  = tid & 31;
  int wv    = tid >> 5;
  int mrow  = lane & 15;
  int khalf = lane >> 4;

  const float* Rb = R + b * (HY * WY);
  for (int e = tid; e < 10 * 144; e += 128) {
    int rr = e / 144, cc = e % 144;
    Rt[e] = (rr < 9 && cc < WY) ? Rb[(i + rr) * WY + cc] : 0.f;
  }
  for (int e = tid; e < 4 * 84 * 16; e += 128) Wsh[e] = Wp[e];
  __syncthreads();

  const int wbase = wv * 1344 + khalf * 32 + mrow;  // (wv*84 + ka)*16 + m
  for (int j0t = 0; j0t < 8; ++j0t) {
    int j0 = j0t * 16;
    int bcol = j0 + mrow;
    v8f acc = {0.f,0.f,0.f,0.f,0.f,0.f,0.f,0.f};
    #pragma unroll
    for (int s = 0; s < 21; ++s) {
      const int k0 = 4 * s, k1 = 4 * s + 1, k2 = 4 * s + 2, k3 = 4 * s + 3;
      const int I0 = (k0 < 81) ? ((k0 / 9) * 144 + (k0 % 9)) : 9 * 144;
      const int I1 = (k1 < 81) ? ((k1 / 9) * 144 + (k1 % 9)) : 9 * 144;
      const int I2 = (k2 < 81) ? ((k2 / 9) * 144 + (k2 % 9)) : 9 * 144;
      const int I3 = (k3 < 81) ? ((k3 / 9) * 144 + (k3 % 9)) : 9 * 144;
      int ia = khalf ? I2 : I0;
      int ib = khalf ? I3 : I1;
      v2f Av, Bv;
      Av.x = Wsh[wbase + s * 64];
      Av.y = Wsh[wbase + s * 64 + 16];
      Bv.x = Rt[ia + bcol];
      Bv.y = Rt[ib + bcol];
      acc = __builtin_amdgcn_wmma_f32_16x16x4_f32(false, Av, false, Bv,
                                                  (short)0, acc, false, false);
    }
    int j = j0 + mrow;
    if (j < WX) {
      #pragma unroll
      for (int r = 0; r < 8; ++r) {
        int a = wv * 16 + r + khalf * 8;   // C layout: M = r (+8 upper lanes)
        g[((b * NA + a) * HX + i) * WX + j] = acc[r];
      }
    }
  }
}

// ---------------- D-cost: partial sumsq(Y - D(soft(X+alpha*g))) ---------------
// grid (4 x0-pairs, 8 y0, 8 b) = 256 tiles; block 256 = 8 waves; each wave
// handles 8 atoms and TWO 16x16 output tiles (x0, x0+16) sharing B fetches.
__global__ __launch_bounds__(256) void dcost_kernel(const float* __restrict__ Xin,
                                                    const float* __restrict__ g,
                                                    const float* __restrict__ Btp,
                                                    const float* __restrict__ Yim,
                                                    float alpha, float talpha,
                                                    int writeR,
                                                    float* __restrict__ Rout,
                                                    float* __restrict__ partial) {
  __shared__ float patch[8][24 * 40];     // per-wave 24x40 padded soft patch
  __shared__ float redbuf[8][2][8][32];   // cross-wave C-tile reduction

  int x0 = blockIdx.x * 32;
  int y0 = blockIdx.y * 16;
  int b  = blockIdx.z;
  int tid   = threadIdx.x;
  int lane  = tid & 31;
  int wv    = tid >> 5;
  int mrow  = lane & 15;
  int khalf = lane >> 4;

  v8f acc0 = {0.f,0.f,0.f,0.f,0.f,0.f,0.f,0.f};
  v8f acc1 = {0.f,0.f,0.f,0.f,0.f,0.f,0.f,0.f};
  const int abase = mrow * 40 + khalf * 2;   // A row base for this lane

  for (int ai = 0; ai < 8; ++ai) {
    int a = wv * 8 + ai;
    const float* Xa = Xin + (size_t)(b * NA + a) * (HX * WX);
    const float* ga = g   + (size_t)(b * NA + a) * (HX * WX);
    if (ai + 1 < 8) {                              // hint next atom into cache
      __builtin_prefetch(Xa + HX * WX + lane * 64, 0, 0);
      __builtin_prefetch(ga + HX * WX + lane * 64, 0, 0);
    }
    // 24x40 patch: Xpad rows y0-8..y0+15, cols x0-8..x0+31 (30 elems/lane)
    for (int e = lane; e < 960; e += 32) {
      int rr = e / 40, cc = e % 40;
      int row = y0 - 8 + rr, col = x0 - 8 + cc;
      float v = 0.f;
      if (row >= 0 && row < HX && col >= 0 && col < WX) {
        int idx = row * WX + col;
        float xv = Xa[idx];
        if (alpha != 0.f) {
          xv += alpha * ga[idx];
          float t = fabsf(xv) - talpha;
          v = (t > 0.f) ? copysignf(t, xv) : 0.f;
        } else {
          v = xv;
        }
      }
      patch[wv][e] = v;
    }
    __syncthreads();
    #pragma unroll
    for (int u = 0; u < 9; ++u) {
      #pragma unroll
      for (int c = 0; c < 6; ++c) {
        int off = abase + u * 40 + 4 * c;
        v2f Av0, Av1, Bv;
        Av0.x = patch[wv][off];
        Av0.y = patch[wv][off + 1];
        Av1.x = patch[wv][off + 16];
        Av1.y = patch[wv][off + 17];
        const float* Bb = Btp + ((a * 9 + u) * 6 + c) * 64;
        Bv.x = Bb[lane];
        Bv.y = Bb[32 + lane];
        acc0 = __builtin_amdgcn_wmma_f32_16x16x4_f32(false, Av0, false, Bv,
                                                     (short)0, acc0, false, false);
        acc1 = __builtin_amdgcn_wmma_f32_16x16x4_f32(false, Av1, false, Bv,
                                                     (short)0, acc1, false, false);
      }
    }
    __syncthreads();
  }

  #pragma unroll
  for (int r = 0; r < 8; ++r) {
    redbuf[wv][0][r][lane] = acc0[r];
    redbuf[wv][1][r][lane] = acc1[r];
  }
  __syncthreads();

  // combine the 8 per-wave partial C tiles, residual vs Y (512 px, 2 per thread)
  float sq = 0.f;
  #pragma unroll
  for (int h = 0; h < 2; ++h) {
    int p = tid + h * 256;
    int tile = p >> 8;
    int idx  = p & 255;
    int r = idx >> 5, l = idx & 31;
    float d = 0.f;
    #pragma unroll
    for (int w = 0; w < 8; ++w) d += redbuf[w][tile][r][l];
    int y = y0 + r + ((l >> 4) << 3);
    int x = x0 + tile * 16 + (l & 15);
    int yidx = (b * HY + y) * WY + x;
    float resid = Yim[yidx] - d;
    if (writeR) Rout[yidx] = resid;
    sq += resid * resid;
  }

  float* red2 = &patch[0][0];
  __syncthreads();
  red2[tid] = sq;
  __syncthreads();
  for (int s = 128; s > 0; s >>= 1) {
    if (tid < s) red2[tid] += red2[tid + s];
    __syncthreads();
  }
  if (tid == 0) {
    int tileIdx = (blockIdx.z * gridDim.y + blockIdx.y) * gridDim.x + blockIdx.x;
    partial[tileIdx] = red2[0];
  }
}

// ---------------- L1 reduce: sum |soft(X + alpha g, talpha)| ------------------
__global__ __launch_bounds__(256) void l1_kernel(const float* __restrict__ X,
                                                 const float* __restrict__ g,
                                                 float alpha, float talpha,
                                                 float* __restrict__ partial, int n) {
  __shared__ float red[256];
  float s = 0.f;
  for (int i = blockIdx.x * 256 + threadIdx.x; i < n; i += gridDim.x * 256) {
    float xv = X[i], v;
    if (alpha != 0.f) {
      xv += alpha * g[i];
      float t = fabsf(xv) - talpha;
      v = (t > 0.f) ? t : 0.f;          // |soft(x,a)| = max(|x|-a, 0)
    } else {
      v = fabsf(xv);
    }
    s += v;
  }
  red[threadIdx.x] = s;
  __syncthreads();
  for (int st = 128; st > 0; st >>= 1) {
    if (threadIdx.x < st) red[threadIdx.x] += red[threadIdx.x + st];
    __syncthreads();
  }
  if (threadIdx.x == 0) partial[blockIdx.x] = red[0];
}

// ---------------- select alpha (scan semantics of the reference) --------------
// psq: 18 slots * 256  (0..16 candidates, 17 = R pass)
// pl1: 17 slots * 1024 (0 = |X|, 1..16 = candidate k)
__global__ void select_kernel(const float* __restrict__ psq,
                              const float* __restrict__ pl1,
                              float* __restrict__ alpha_out) {
  float sumsqR = 0.f;
  for (int i = 0; i < DTILES; ++i) sumsqR += psq[17 * DTILES + i];
  float l1X = 0.f;
  for (int i = 0; i < L1BLOCKS; ++i) l1X += pl1[i];
  float current = sumsqR + TAUC * l1X;
  int sel = 16;
  for (int k = 0; k < 16; ++k) {
    float sq = 0.f;
    for (int i = 0; i < DTILES; ++i) sq += psq[k * DTILES + i];
    float l1 = l1X;                         // cost0 uses l1 of X (faithful)
    if (k > 0) {
      l1 = 0.f;
      for (int i = 0; i < L1BLOCKS; ++i) l1 += pl1[k * L1BLOCKS + i];
    }
    if (sq + TAUC * l1 < current) { sel = k; break; }
  }
  alpha_out[0] = ldexpf(1.f, -sel);
}

// ---------------- apply selected update ---------------------------------------
__global__ __launch_bounds__(256) void update_kernel(const float* __restrict__ X,
                                                     const float* __restrict__ g,
                                                     const float* __restrict__ alpha_p,
                                                     float* __restrict__ out, int n) {
  float alpha = alpha_p[0];
  float ta = TAUC * alpha;
  for (int i = blockIdx.x * 256 + threadIdx.x; i < n; i += gridDim.x * 256) {
    float xv = X[i] + alpha * g[i];
    float t = fabsf(xv) - ta;
    out[i] = (t > 0.f) ? copysignf(t, xv) : 0.f;
  }
}

// ---------------- FISTA momentum ----------------------------------------------
__global__ __launch_bounds__(256) void momentum_kernel(const float* __restrict__ X2,
                                                       const float* __restrict__ X1p,
                                                       float cf, float* __restrict__ Z,
                                                       int n) {
  for (int i = blockIdx.x * 256 + threadIdx.x; i < n; i += gridDim.x * 256) {
    float a = X2[i];
    Z[i] = a + (a - X1p[i]) * cf;
  }
}

// ---------------- host orchestration ------------------------------------------
extern "C" void kernel_launch(void* const* d_in, const int* in_sizes, int n_in,
                              void* d_out, int out_size, void* d_ws, size_t ws_size,
                              hipStream_t stream) {
  (void)in_sizes; (void)n_in; (void)out_size; (void)ws_size;
  const float* Yim  = (const float*)d_in[0];
  const float* X1in = (const float*)d_in[1];
  const float* Wd   = (const float*)d_in[2];

  // carve workspace (floats, 256B aligned slices); total ~119.5 MB
  size_t off = 0;
  auto carve = [&](size_t nfl) {
    float* p = (float*)d_ws + off;
    off += (nfl + 63) & ~(size_t)63;
    return p;
  };
  float* Wp    = carve(4 * 84 * 16);
  float* Btp   = carve(64 * 9 * 6 * 64);
  float* Rbuf  = carve(NYEL);
  float* gbuf  = carve(NXEL);
  float* bufA  = carve(NXEL);
  float* bufB  = carve(NXEL);
  float* bufZ  = carve(NXEL);
  float* psq   = carve(18 * DTILES);
  float* pl1   = carve(17 * L1BLOCKS);
  float* alphp = carve(16);

  prep_dt_kernel<<<(4 * 84 * 16 + 255) / 256, 256, 0, stream>>>(Wd, Wp);
  prep_toep_kernel<<<(64 * 9 * 6 * 64) / 256, 256, 0, stream>>>(Wd, Btp);

  float alphas[17];
  for (int k = 0; k < 17; ++k) alphas[k] = (float)ldexp(1.0, -k);

  dim3 dgrid(4, 8, 8);      // x0-pairs, y0, b
  dim3 tgrid(HX, NB);       // row i, b

  auto linesearch = [&](const float* Xls, float* dst) {
    // R = Y - D(X)  (alpha=0 pass) + current-cost pieces
    dcost_kernel<<<dgrid, 256, 0, stream>>>(Xls, gbuf, Btp, Yim, 0.f, 0.f, 1,
                                            Rbuf, psq + 17 * DTILES);
    l1_kernel<<<L1BLOCKS, 256, 0, stream>>>(Xls, gbuf, 0.f, 0.f, pl1, NXEL);
    // g = Dt(R)
    dt_wmma_kernel<<<tgrid, 128, 0, stream>>>(Rbuf, Wp, gbuf);
    // candidate costs k = 0..16 (alpha = 2^-k); cost0 uses l1(X)
    for (int k = 1; k <= 16; ++k)
      l1_kernel<<<L1BLOCKS, 256, 0, stream>>>(Xls, gbuf, alphas[k],
                                              TAUC * alphas[k],
                                              pl1 + k * L1BLOCKS, NXEL);
    for (int k = 0; k <= 16; ++k)
      dcost_kernel<<<dgrid, 256, 0, stream>>>(Xls, gbuf, Btp, Yim, alphas[k],
                                              TAUC * alphas[k], 0, nullptr,
                                              psq + k * DTILES);
    select_kernel<<<1, 1, 0, stream>>>(psq, pl1, alphp);
    update_kernel<<<2048, 256, 0, stream>>>(Xls, gbuf, alphp, dst, NXEL);
  };

  // X2 = LS(X1)
  linesearch(X1in, bufA);
  double t1 = 1.0, t2;

  // iter 0: t1=1 -> momentum coefficient 0 -> Z = X2 exactly
  t2 = (1.0 + sqrt(1.0 + 4.0 * t1 * t1)) / 2.0;
  linesearch(bufA, bufB);
  t1 = t2;

  // iter 1
  t2 = (1.0 + sqrt(1.0 + 4.0 * t1 * t1)) / 2.0;
  momentum_kernel<<<2048, 256, 0, stream>>>(bufB, bufA,
                                            (float)((t1 - 1.0) / t2), bufZ, NXEL);
  linesearch(bufZ, bufA);
  t1 = t2;

  // iter 2 (final -> d_out)
  t2 = (1.0 + sqrt(1.0 + 4.0 * t1 * t1)) / 2.0;
  momentum_kernel<<<2048, 256, 0, stream>>>(bufA, bufB,
                                            (float)((t1 - 1.0) / t2), bufZ, NXEL);
  linesearch(bufZ, (float*)d_out);
}